// AttentionPooling_18700287607677
// MI455X (gfx1250) — compile-verified
//
#include <hip/hip_runtime.h>
#include <hip/hip_bf16.h>

// Problem constants (match reference)
#define BB 64
#define SS 4096
#define DD 256
#define LN_EPS 1e-3f
#define NSPLIT 8             // split-K over sequence for pooling
#define SCHUNK (SS / NSPLIT) // 512

typedef __attribute__((ext_vector_type(2))) float v2f;
typedef __attribute__((ext_vector_type(8))) float v8f;

// ---------------- Kernel 0: gw = gamma*w, sgw = sum(gw), c0 = sum(beta*w) ----
__global__ void ap_prep_kernel(const float* __restrict__ gamma,
                               const float* __restrict__ beta,
                               const float* __restrict__ w,
                               float* __restrict__ gw,
                               float* __restrict__ scal) {
    __shared__ float r1[DD];
    __shared__ float r2[DD];
    int d = threadIdx.x;
    float g = gamma[d] * w[d];
    gw[d] = g;
    r1[d] = g;
    r2[d] = beta[d] * w[d];
    __syncthreads();
    for (int off = DD / 2; off > 0; off >>= 1) {
        if (d < off) { r1[d] += r1[d + off]; r2[d] += r2[d + off]; }
        __syncthreads();
    }
    if (d == 0) { scal[0] = r1[0]; scal[1] = r2[0]; }
}

// ---------------- Kernel 1: fused LayerNorm + Dense(1) scores -----------------
// One wave32 per (b,s) row of 256 floats. 8 rows per 256-thread block.
// score = rstd*(dot(x,gamma*w) - mu*sum(gamma*w)) + dot(beta,w) + bias
__global__ void ap_scores_kernel(const float* __restrict__ x,
                                 const float* __restrict__ mask,
                                 const float* __restrict__ gwbuf,
                                 const float* __restrict__ scal,
                                 const float* __restrict__ bias,
                                 float* __restrict__ scores) {
    __shared__ __align__(16) float gws[DD];
    int tid = threadIdx.x;
    gws[tid] = gwbuf[tid];
    __syncthreads();

    int wid  = tid >> 5;
    int lane = tid & 31;
    long row = (long)blockIdx.x * 8 + wid;   // 0 .. B*S-1

    const float4* rp = (const float4*)(x + row * DD);
    float4 v0 = rp[lane];
    float4 v1 = rp[lane + 32];
    const float4* gp = (const float4*)gws;
    float4 g0 = gp[lane];
    float4 g1 = gp[lane + 32];

    float s1 = (v0.x + v0.y + v0.z + v0.w) + (v1.x + v1.y + v1.z + v1.w);
    float s2 = v0.x*v0.x + v0.y*v0.y + v0.z*v0.z + v0.w*v0.w
             + v1.x*v1.x + v1.y*v1.y + v1.z*v1.z + v1.w*v1.w;
    float sd = v0.x*g0.x + v0.y*g0.y + v0.z*g0.z + v0.w*g0.w
             + v1.x*g1.x + v1.y*g1.y + v1.z*g1.z + v1.w*g1.w;

    // wave32 tree reduction
    for (int off = 16; off > 0; off >>= 1) {
        s1 += __shfl_xor(s1, off, 32);
        s2 += __shfl_xor(s2, off, 32);
        sd += __shfl_xor(sd, off, 32);
    }
    if (lane == 0) {
        float mu   = s1 * (1.0f / DD);
        float var  = s2 * (1.0f / DD) - mu * mu;
        float rstd = rsqrtf(var + LN_EPS);
        float sc   = rstd * (sd - mu * scal[0]) + scal[1] + bias[0];
        sc += (1.0f - mask[row]) * -1e9f;
        scores[row] = sc;
    }
}

// ---------------- Kernel 2: softmax over S per batch row ----------------------
__global__ void ap_softmax_kernel(float* __restrict__ scores) {
    __shared__ float red[256];
    __shared__ float bcast;
    int b = blockIdx.x, tid = threadIdx.x;
    float* row = scores + (size_t)b * SS;

    float m = -INFINITY;
    for (int s = tid; s < SS; s += 256) m = fmaxf(m, row[s]);
    red[tid] = m; __syncthreads();
    for (int off = 128; off > 0; off >>= 1) {
        if (tid < off) red[tid] = fmaxf(red[tid], red[tid + off]);
        __syncthreads();
    }
    if (tid == 0) bcast = red[0];
    __syncthreads();
    float mm = bcast;
    __syncthreads();

    float z = 0.0f;
    for (int s = tid; s < SS; s += 256) {
        float e = __expf(row[s] - mm);
        row[s] = e;
        z += e;
    }
    red[tid] = z; __syncthreads();
    for (int off = 128; off > 0; off >>= 1) {
        if (tid < off) red[tid] += red[tid + off];
        __syncthreads();
    }
    if (tid == 0) bcast = red[0];
    __syncthreads();
    float inv = 1.0f / bcast;
    for (int s = tid; s < SS; s += 256) row[s] *= inv;
}

// ---------------- Kernel 3: weighted pooling via V_WMMA_F32_16X16X4_F32 -------
// One wave per (split, b, d-tile). A = softmax weights broadcast over M rows,
// B = x tile (k = s, n = d), C chained across 128 WMMAs (K = 512 per wave).
// A 16x4 f32 layout: lane L -> m = L%16; VGPR0 holds k = (L<16?0:2), VGPR1 k+1.
// B 4x16 f32 layout: lane L -> n = L%16; VGPR0 holds k = (L<16?0:2), VGPR1 k+1.
// D 16x16 f32: element (0, n) lives in C VGPR0 on lanes 0-15.
// Pointer-increment inner loop: constant 4-row (4 KB) stride folds into the
// 24-bit immediate offset field under unroll-8 (one u64 add per 8 WMMAs).
__global__ void ap_pool_wmma_kernel(const float* __restrict__ x,
                                    const float* __restrict__ wts,
                                    float* __restrict__ part) {
    __shared__ __align__(16) float wsh[SCHUNK];
    int lane  = threadIdx.x;
    int id    = blockIdx.x;
    int dt    = id & 15;           // d-tile: 0..15
    int b     = (id >> 4) & 63;    // batch
    int split = id >> 10;          // split-K chunk: 0..7
    int s0    = split * SCHUNK;

    const float* wrow = wts + (size_t)b * SS + s0;
    for (int i = lane; i < SCHUNK; i += 32) wsh[i] = wrow[i];
    __syncthreads();

    int half = lane >> 4;          // 0 for lanes 0-15, 1 for 16-31
    int kb   = half * 2;           // this half's K base within the 4-slice
    int dd   = dt * 16 + (lane & 15);

    const float* p  = x + ((size_t)b * SS + s0 + kb) * DD + dd;
    const float* wp = wsh + kb;

    v8f c = {0.f, 0.f, 0.f, 0.f, 0.f, 0.f, 0.f, 0.f};
    #pragma unroll 8
    for (int it = 0; it < SCHUNK / 4; ++it) {
        float b0 = p[0];           // x[b, s+kb,   dd]
        float b1 = p[DD];          // x[b, s+kb+1, dd]
        v2f A  = { wp[0], wp[1] };                    // weights on A (any m)
        v2f Bv = { b0, b1 };                          // x on B (n = d)
        c = __builtin_amdgcn_wmma_f32_16x16x4_f32(
                /*neg_a=*/false, A, /*neg_b=*/false, Bv,
                /*c_mod=*/(short)0, c, /*reuse_a=*/false, /*reuse_b=*/false);
        p  += 4 * DD;
        wp += 4;
    }
    if (lane < 16) {
        part[(((size_t)split * BB + b) * DD) + dt * 16 + (lane & 15)] = c[0];
    }
}

// ---------------- Kernel 4: deterministic split-K reduction -------------------
__global__ void ap_reduce_kernel(const float* __restrict__ part,
                                 float* __restrict__ out) {
    int i = blockIdx.x * 256 + threadIdx.x;   // 0 .. B*D-1
    float acc = 0.0f;
    #pragma unroll
    for (int k = 0; k < NSPLIT; ++k) acc += part[(size_t)k * (BB * DD) + i];
    out[i] = acc;
}

extern "C" void kernel_launch(void* const* d_in, const int* in_sizes, int n_in,
                              void* d_out, int out_size, void* d_ws, size_t ws_size,
                              hipStream_t stream) {
    const float* x     = (const float*)d_in[0];
    const float* mask  = (const float*)d_in[1];
    const float* gamma = (const float*)d_in[2];
    const float* beta  = (const float*)d_in[3];
    const float* w     = (const float*)d_in[4];
    const float* bias  = (const float*)d_in[5];
    float* out = (float*)d_out;

    float* ws     = (float*)d_ws;
    float* gw     = ws;                       // 256 floats
    float* scal   = ws + 256;                 // 2 floats (+pad)
    float* scores = ws + 512;                 // B*S = 262144 floats (1 MB)
    float* part   = ws + 512 + BB * SS;       // NSPLIT*B*D = 131072 floats (512 KB)

    ap_prep_kernel<<<1, DD, 0, stream>>>(gamma, beta, w, gw, scal);
    ap_scores_kernel<<<(BB * SS) / 8, 256, 0, stream>>>(x, mask, gw, scal, bias, scores);
    ap_softmax_kernel<<<BB, 256, 0, stream>>>(scores);
    ap_pool_wmma_kernel<<<NSPLIT * BB * (DD / 16), 32, 0, stream>>>(x, scores, part);
    ap_reduce_kernel<<<(BB * DD) / 256, 256, 0, stream>>>(part, out);
}